// MembraneLayer_72292889526609
// MI455X (gfx1250) — compile-verified
//
#include <hip/hip_runtime.h>
#include <stdint.h>

typedef __attribute__((ext_vector_type(2))) float v2f;
typedef __attribute__((ext_vector_type(8))) float v8f;
typedef __attribute__((ext_vector_type(4))) unsigned int v4u;
typedef __attribute__((ext_vector_type(8))) int v8i;
typedef __attribute__((ext_vector_type(4))) int v4i;

#define Bq 512
#define Tq 100
#define Cq 700
#define Dq 512
#define Mq (Bq * Tq)   // 51200 rows

#define LDA 34       // A LDS row stride (32 data + 2 pad) == TDM pad_interval/amount
#define BBLK 288     // floats per B k-pair block: 128 pairs + 32 pad (half-wave bank split)
#define ABUF (64 * LDA)    // 2176 floats per A buffer
#define BBUF (16 * BBLK)   // 4608 floats per B buffer
#define NCHUNK 22          // ceil(700 / 32)
#define SMEM_FLOATS (2 * ABUF + 2 * BBUF)   // 13568 floats = 54272 B

// Issue one TDM DMA: 64x32 f32 tile of A starting at (m0, k0) -> LDS (stride-34 rows).
// D# per CDNA5 ISA 8.3/8.4. tensor_dim0 = remaining K extent so the hardware
// zero-fills the k>=700 tail. pad: every 32 DWORDs insert 2 DWORDs -> row stride 34.
__device__ __forceinline__ void tdm_load_A(const float* gA, unsigned lds_byte,
                                           int m0, int k0) {
  uint64_t ga = (uint64_t)(uintptr_t)gA + ((uint64_t)m0 * Cq + (uint64_t)k0) * 4ull;
  unsigned dim0 = (unsigned)(Cq - k0);   // remaining valid K from tile start
  v4u g0;
  g0.x = 1u;                                             // count=1 (valid user D#)
  g0.y = lds_byte;                                       // lds_addr
  g0.z = (unsigned)(ga & 0xFFFFFFFFu);                   // global_addr[31:0]
  g0.w = (unsigned)((ga >> 32) & 0x01FFFFFFu) | (2u << 30);  // addr[56:32] | type=2
  v8i g1;
  g1[0] = (int)((2u << 16)        // data_size = 4 bytes
              | (1u << 20)        // pad_enable
              | (4u << 22)        // pad_interval: 32 DWORDs
              | (1u << 25));      // pad_amount: 2 DWORDs
  g1[1] = (int)((dim0 & 0xFFFFu) << 16);                     // tensor_dim0[15:0]
  g1[2] = (int)(((dim0 >> 16) & 0xFFFFu) | (64u << 16));     // dim0 hi | tensor_dim1=64
  g1[3] = (int)(32u << 16);                                  // tile_dim0 = 32
  g1[4] = 64;                                                // tile_dim1 = 64
  g1[5] = Cq;                                                // tensor_dim0_stride lo
  g1[6] = 0;                                                 // stride hi | dim1_stride lo
  g1[7] = 0;
  v4i zz4 = {0, 0, 0, 0};              // groups 2/3 unused (2-D tensor)
  v8i zz8 = {0, 0, 0, 0, 0, 0, 0, 0};  // 6-arg toolchain form: extra group
  __builtin_amdgcn_tensor_load_to_lds(g0, g1, zz4, zz4, zz8, 0);
}

// Stage W tile (32 k x 128 n) as interleaved (2q,2q+1) row pairs -> fragment-ready b64s.
__device__ __forceinline__ void stage_W(const float* __restrict__ W, float* dst,
                                        int k0, int n0, int tid) {
  for (int t = tid; t < 512; t += 256) {
    int q  = t >> 5;            // 0..15 -> global k rows 2q, 2q+1
    int nq = (t & 31) << 2;     // column quad 0,4,...,124
    int gk = k0 + (q << 1);     // even; gk < 700 implies gk+1 <= 699
    float4 r0 = make_float4(0.f, 0.f, 0.f, 0.f);
    float4 r1 = r0;
    if (gk < Cq) {
      r0 = *reinterpret_cast<const float4*>(W + (size_t)gk * Dq + n0 + nq);
      r1 = *reinterpret_cast<const float4*>(W + (size_t)(gk + 1) * Dq + n0 + nq);
    }
    v2f* p = reinterpret_cast<v2f*>(dst + q * BBLK + (nq << 1));
    v2f p0; p0.x = r0.x; p0.y = r1.x;
    v2f p1; p1.x = r0.y; p1.y = r1.y;
    v2f p2; p2.x = r0.z; p2.y = r1.z;
    v2f p3; p3.x = r0.w; p3.y = r1.w;
    p[0] = p0; p[1] = p1; p[2] = p2; p[3] = p3;
  }
}

// H[M, D] = inputs[M, C] * w[C, D] via V_WMMA_F32_16X16X4_F32.
// 256 threads = 8 waves; block tile 64(M) x 128(N); K chunk 32, double-buffered:
// A chunks arrive by TDM (tensor_load_to_lds, TENSORcnt), W staged by all threads;
// chunk k+1's DMA/loads overlap chunk k's 32 WMMAs.
__global__ __launch_bounds__(256)
void membrane_gemm_wmma(const float* __restrict__ A, const float* __restrict__ W,
                        float* __restrict__ H) {
  extern __shared__ float smem[];
  float* Alds = smem;                 // 2 buffers: [0, 2*ABUF)
  float* Blds = smem + 2 * ABUF;      // 2 buffers: [2*ABUF, 2*ABUF + 2*BBUF)

  const int tid   = threadIdx.x;
  const int lane  = tid & 31;
  const int wave  = tid >> 5;
  const int waveM = wave >> 1;    // 0..3 -> 16-row strip
  const int waveN = wave & 1;     // 0..1 -> 64-col strip
  const int m0 = blockIdx.y * 64;
  const int n0 = blockIdx.x * 128;

  const int l15 = lane & 15;
  const int h   = lane >> 4;      // ISA frag: lanes 16-31 hold K+2,K+3

  v8f acc[4] = {};

  const float* Abase = Alds + (waveM * 16 + l15) * LDA + (h << 1);
  const float* Bbase = Blds + h * BBLK + ((waveN * 64 + l15) << 1);

  // Prologue: stage chunk 0 into buffer 0
  stage_W(W, Blds, 0, n0, tid);
  if (wave == 0) tdm_load_A(A, 0u, m0, 0);

  for (int kc = 0; kc < NCHUNK; ++kc) {
    const int cur = kc & 1;
    const int nxt = cur ^ 1;
    if (wave == 0) __builtin_amdgcn_s_wait_tensorcnt(0);  // A[cur] DMA complete
    __syncthreads();                                      // publish buffers[cur]
    if (kc + 1 < NCHUNK) {                                // prefetch chunk kc+1
      stage_W(W, Blds + nxt * BBUF, (kc + 1) * 32, n0, tid);
      if (wave == 0) tdm_load_A(A, (unsigned)(nxt * ABUF * 4), m0, (kc + 1) * 32);
    }
    const float* Ab = Abase + cur * ABUF;
    const float* Bb = Bbase + cur * BBUF;
#pragma unroll
    for (int g = 0; g < 8; ++g) {          // 8 WMMA K-steps of 4
      v2f a = *reinterpret_cast<const v2f*>(Ab + (g << 2));
#pragma unroll
      for (int nt = 0; nt < 4; ++nt) {
        v2f b = *reinterpret_cast<const v2f*>(Bb + g * (2 * BBLK) + (nt << 5));
        acc[nt] = __builtin_amdgcn_wmma_f32_16x16x4_f32(
            /*neg_a=*/false, a, /*neg_b=*/false, b,
            /*c_mod=*/(short)0, acc[nt], /*reuse_a=*/false, /*reuse_b=*/false);
      }
    }
    __syncthreads();   // done reading buffers[cur] before it is restaged
  }

  // C/D layout: VGPR v -> M = v (lanes 0-15) / 8+v (lanes 16-31), N = lane&15
  const int mrow0 = m0 + waveM * 16 + (h << 3);
#pragma unroll
  for (int nt = 0; nt < 4; ++nt) {
    int ncol = n0 + waveN * 64 + nt * 16 + l15;
#pragma unroll
    for (int v = 0; v < 8; ++v) {
      H[(size_t)(mrow0 + v) * Dq + ncol] = acc[nt][v];
    }
  }
}

// Leaky-integrator scan: one thread per (b, d); consecutive threads walk d ->
// fully coalesced. State in registers across T; matches torch loop order.
__global__ __launch_bounds__(256)
void membrane_scan(const float* __restrict__ H, const float* __restrict__ alpha,
                   const float* __restrict__ beta, float* __restrict__ outS,
                   float* __restrict__ outM) {
  int tid = blockIdx.x * 256 + threadIdx.x;   // 0 .. B*D-1
  int b = tid >> 9;          // / D (D = 512)
  int d = tid & (Dq - 1);
  float al = alpha[d];
  float be = beta[d];
  float omb = 1.0f - be;

  size_t base = (size_t)b * Tq * Dq + d;
  float syn = 0.f, mem = 0.f;
  outS[base] = 0.f;          // reference records the zero initial state at t=0
  outM[base] = 0.f;
  for (int s = 0; s < Tq - 1; ++s) {
    float hh = H[base + (size_t)s * Dq];
    float nm = be * mem + omb * syn;   // uses PRE-update syn
    syn = al * syn + hh;
    mem = nm;
    size_t o = base + (size_t)(s + 1) * Dq;
    outS[o] = syn;
    outM[o] = mem;
  }
}

extern "C" void kernel_launch(void* const* d_in, const int* in_sizes, int n_in,
                              void* d_out, int out_size, void* d_ws, size_t ws_size,
                              hipStream_t stream) {
  const float* inputs = (const float*)d_in[0];  // [B, T, C]
  const float* w      = (const float*)d_in[1];  // [C, D]
  const float* alpha  = (const float*)d_in[2];  // [1, D]
  const float* beta   = (const float*)d_in[3];  // [1, D]

  float* H    = (float*)d_ws;                   // [B*T, D] intermediate (100 MB)
  float* outS = (float*)d_out;                  // syn_rec [B, T, D]
  float* outM = outS + (size_t)Bq * Tq * Dq;    // mem_rec [B, T, D]

  dim3 grid(Dq / 128, Mq / 64);                 // (4, 800)
  membrane_gemm_wmma<<<grid, 256, SMEM_FLOATS * sizeof(float), stream>>>(inputs, w, H);
  membrane_scan<<<(Bq * Dq) / 256, 256, 0, stream>>>(H, alpha, beta, outS, outM);
}